// BaseGNN_13589276524898
// MI455X (gfx1250) — compile-verified
//
#include <hip/hip_runtime.h>
#include <cstdint>
#include <cstddef>

typedef float v2f __attribute__((ext_vector_type(2)));
typedef float v8f __attribute__((ext_vector_type(8)));

#define H_DIM 128
#define C_DIM 64
#define BN_EPS 1e-5f

__device__ __forceinline__ void atomAddF(float* p, float v) {
    // hardware global_atomic_add_f32 (no CAS loop)
    unsafeAtomicAdd(p, v);
}

// ---------------- degree / dinv ----------------
__global__ void k_init_deg(float* __restrict__ deg, int n) {
    int i = blockIdx.x * blockDim.x + threadIdx.x;
    if (i < n) deg[i] = 1.0f;  // self loop contributes 1
}

__global__ void k_deg_accum(const int* __restrict__ dst, float* __restrict__ deg, int e) {
    int i = blockIdx.x * blockDim.x + threadIdx.x;
    int stride = gridDim.x * blockDim.x;
    for (; i < e; i += stride) atomAddF(&deg[dst[i]], 1.0f);
}

__global__ void k_deg_to_dinv(float* __restrict__ deg, int n) {
    int i = blockIdx.x * blockDim.x + threadIdx.x;
    if (i < n) deg[i] = rsqrtf(deg[i]);  // deg >= 1 always
}

// ---------------- WMMA f32 GEMM: Y[n x OUTC] = X[n x 128] @ W[128 x OUTC] (+bias) ----------
template <int OUTC, int WAVES>
__global__ __launch_bounds__(32 * WAVES)
void k_gemm_wmma(const float* __restrict__ X, const float* __restrict__ W,
                 const float* __restrict__ bias, float* __restrict__ Y, int n) {
    __shared__ float As[16 * H_DIM];  // 8KB row strip
    const int r0  = blockIdx.x * 16;
    const int tid = threadIdx.y * 32 + threadIdx.x;
    for (int i = tid; i < 16 * H_DIM; i += 32 * WAVES) {
        int r = r0 + (i >> 7);
        As[i] = (r < n) ? X[(size_t)r * H_DIM + (i & 127)] : 0.0f;
    }
    __syncthreads();

    const int lane = threadIdx.x;
    const int lm   = lane & 15;      // M index (A) / N index (B,D)
    const int kg   = lane >> 4;      // K-group: lanes 16-31 carry K+2,K+3 / rows M+8 of D
    const int c0   = threadIdx.y * 16;

    v8f acc = {};
#pragma unroll
    for (int kb = 0; kb < H_DIM / 4; ++kb) {
        const int k = kb * 4 + kg * 2;
        v2f a, b;
        a.x = As[lm * H_DIM + k];
        a.y = As[lm * H_DIM + k + 1];
        b.x = W[(size_t)k * OUTC + c0 + lm];
        b.y = W[(size_t)(k + 1) * OUTC + c0 + lm];
        acc = __builtin_amdgcn_wmma_f32_16x16x4_f32(false, a, false, b,
                                                    (short)0, acc, false, false);
    }

    const float bv = bias ? bias[c0 + lm] : 0.0f;
#pragma unroll
    for (int v = 0; v < 8; ++v) {
        int r = r0 + v + kg * 8;
        if (r < n) Y[(size_t)r * OUTC + c0 + lm] = acc[v] + bv;
    }
}

// ---------------- aggregation: agg = b + dinv^2 * h (self loop), then edge scatter ----------
__global__ void k_agg_init(const float* __restrict__ h, const float* __restrict__ dinv,
                           const float* __restrict__ bias, float* __restrict__ agg, int n) {
    int idx = blockIdx.x * blockDim.x + threadIdx.x;
    if (idx < n * H_DIM) {
        int i = idx >> 7, c = idx & 127;
        float d = dinv[i];
        agg[idx] = bias[c] + d * d * h[idx];
    }
}

__global__ void k_agg_edges(const float* __restrict__ h, const int* __restrict__ src,
                            const int* __restrict__ dst, const float* __restrict__ dinv,
                            float* __restrict__ agg, int e) {
    const int lane    = threadIdx.x & 31;
    int wid           = (blockIdx.x * blockDim.x + threadIdx.x) >> 5;
    const int wstride = (gridDim.x * blockDim.x) >> 5;
    for (int ed = wid; ed < e; ed += wstride) {
        int s = src[ed], d = dst[ed];
        int en = ed + wstride;                 // prefetch next source row
        if (en < e) {
            int sn = src[en];
            __builtin_prefetch(h + (size_t)sn * H_DIM + lane * 4, 0, 0);
        }
        float w = dinv[s] * dinv[d];
        const float4 hv = ((const float4*)(h + (size_t)s * H_DIM))[lane];
        float* ap = agg + (size_t)d * H_DIM + lane * 4;
        atomAddF(ap + 0, hv.x * w);
        atomAddF(ap + 1, hv.y * w);
        atomAddF(ap + 2, hv.z * w);
        atomAddF(ap + 3, hv.w * w);
    }
}

// ---------------- batchnorm ----------------
__global__ void k_bn_zero(float* __restrict__ stats) {
    stats[threadIdx.x] = 0.0f;  // 256 entries: sum[128], sumsq[128]
}

__global__ void k_bn_stats(const float* __restrict__ h, float* __restrict__ stats, int n) {
    int c = threadIdx.x;  // 0..127
    float s = 0.f, sq = 0.f;
    for (int r = blockIdx.x; r < n; r += gridDim.x) {
        float v = h[(size_t)r * H_DIM + c];
        s += v;
        sq += v * v;
    }
    atomAddF(&stats[c], s);
    atomAddF(&stats[H_DIM + c], sq);
}

__global__ void k_bn_final(const float* __restrict__ stats, const float* __restrict__ gamma,
                           const float* __restrict__ beta, float* __restrict__ ss, int n) {
    int c = threadIdx.x;
    if (c < H_DIM) {
        float inv_n = 1.0f / (float)n;
        float mu  = stats[c] * inv_n;
        float var = stats[H_DIM + c] * inv_n - mu * mu;
        float sc  = gamma[c] * rsqrtf(var + BN_EPS);
        ss[c]          = sc;
        ss[H_DIM + c]  = beta[c] - mu * sc;
    }
}

__global__ void k_bn_apply_relu(float* __restrict__ h, const float* __restrict__ ss, int n) {
    int idx = blockIdx.x * blockDim.x + threadIdx.x;
    if (idx < n * H_DIM) {
        int c = idx & 127;
        float v = h[idx] * ss[c] + ss[H_DIM + c];
        h[idx] = fmaxf(v, 0.0f);
    }
}

// ---------------- log-softmax over C=64, one wave32 per row ----------------
__global__ void k_logsoftmax(float* __restrict__ out, int n) {
    int row  = blockIdx.x * (blockDim.x >> 5) + (threadIdx.x >> 5);
    int lane = threadIdx.x & 31;
    if (row >= n) return;
    float2* p = (float2*)(out + (size_t)row * C_DIM);
    float2 z = p[lane];
    float m = fmaxf(z.x, z.y);
    for (int o = 16; o > 0; o >>= 1) m = fmaxf(m, __shfl_xor(m, o, 32));
    float s = expf(z.x - m) + expf(z.y - m);
    for (int o = 16; o > 0; o >>= 1) s += __shfl_xor(s, o, 32);
    float l = m + logf(s);
    p[lane] = make_float2(z.x - l, z.y - l);
}

// ---------------- host launcher ----------------
extern "C" void kernel_launch(void* const* d_in, const int* in_sizes, int n_in,
                              void* d_out, int out_size, void* d_ws, size_t ws_size,
                              hipStream_t stream) {
    const int n = in_sizes[0] / H_DIM;   // 100000
    const int e = in_sizes[1] / 2;       // 1600000

    const float* x   = (const float*)d_in[0];
    const int*   ei  = (const int*)d_in[1];
    const int*   src = ei;
    const int*   dst = ei + e;
    const float* W1  = (const float*)d_in[2];
    const float* b1  = (const float*)d_in[3];
    const float* g1  = (const float*)d_in[4];
    const float* be1 = (const float*)d_in[5];
    const float* W2  = (const float*)d_in[6];
    const float* b2  = (const float*)d_in[7];
    const float* g2  = (const float*)d_in[8];
    const float* be2 = (const float*)d_in[9];
    const float* Wc  = (const float*)d_in[10];
    const float* bc  = (const float*)d_in[11];
    float* out = (float*)d_out;

    // workspace carve-up (256B aligned)
    char* base = (char*)d_ws;
    size_t szDinv = ((size_t)n * 4 + 255) & ~(size_t)255;
    size_t szBuf  = ((size_t)n * H_DIM * 4 + 255) & ~(size_t)255;
    float* dinv  = (float*)base;
    float* buf0  = (float*)(base + szDinv);
    float* buf1  = (float*)(base + szDinv + szBuf);
    float* stats = (float*)(base + szDinv + 2 * szBuf);
    float* ss    = (float*)(base + szDinv + 2 * szBuf + 1024);

    const int gN   = (n + 255) / 256;
    const int gEl  = (n * H_DIM + 255) / 256;
    const int gTil = (n + 15) / 16;
    const dim3 blk8(32, 8), blk4(32, 4);

    // normalization factors (shared by both layers)
    k_init_deg<<<gN, 256, 0, stream>>>(dinv, n);
    k_deg_accum<<<2048, 256, 0, stream>>>(dst, dinv, e);
    k_deg_to_dinv<<<gN, 256, 0, stream>>>(dinv, n);

    // ----- layer 1 -----
    k_gemm_wmma<128, 8><<<gTil, blk8, 0, stream>>>(x, W1, nullptr, buf0, n);
    k_agg_init<<<gEl, 256, 0, stream>>>(buf0, dinv, b1, buf1, n);
    k_agg_edges<<<4096, 256, 0, stream>>>(buf0, src, dst, dinv, buf1, e);
    k_bn_zero<<<1, 256, 0, stream>>>(stats);
    k_bn_stats<<<512, 128, 0, stream>>>(buf1, stats, n);
    k_bn_final<<<1, 128, 0, stream>>>(stats, g1, be1, ss, n);
    k_bn_apply_relu<<<gEl, 256, 0, stream>>>(buf1, ss, n);

    // ----- layer 2 -----
    k_gemm_wmma<128, 8><<<gTil, blk8, 0, stream>>>(buf1, W2, nullptr, buf0, n);
    k_agg_init<<<gEl, 256, 0, stream>>>(buf0, dinv, b2, buf1, n);
    k_agg_edges<<<4096, 256, 0, stream>>>(buf0, src, dst, dinv, buf1, e);
    k_bn_zero<<<1, 256, 0, stream>>>(stats);
    k_bn_stats<<<512, 128, 0, stream>>>(buf1, stats, n);
    k_bn_final<<<1, 128, 0, stream>>>(stats, g2, be2, ss, n);
    k_bn_apply_relu<<<gEl, 256, 0, stream>>>(buf1, ss, n);

    // ----- classifier + log_softmax -----
    k_gemm_wmma<64, 4><<<gTil, blk4, 0, stream>>>(buf1, Wc, bc, out, n);
    k_logsoftmax<<<(n + 7) / 8, 256, 0, stream>>>(out, n);
}